// GCN_41626823032945
// MI455X (gfx1250) — compile-verified
//
#include <hip/hip_runtime.h>

typedef float v2f __attribute__((ext_vector_type(2)));
typedef float v8f __attribute__((ext_vector_type(8)));

// ---------------- degree / norm kernels ----------------
__global__ void k_init_deg(float* deg, int n) {
    int i = blockIdx.x * blockDim.x + threadIdx.x;
    if (i < n) deg[i] = 1.0f;                      // self-loop contributes 1
}

__global__ void k_count_deg(const int* __restrict__ dst, float* deg, int E) {
    int e = blockIdx.x * blockDim.x + threadIdx.x;
    if (e < E) atomicAdd(&deg[dst[e]], 1.0f);
}

__global__ void k_rsqrt_inplace(float* deg, int n) {
    int i = blockIdx.x * blockDim.x + threadIdx.x;
    if (i < n) {
        float d = deg[i];
        deg[i] = (d > 0.0f) ? rsqrtf(d) : 0.0f;
    }
}

// ---------------- fp32 WMMA GEMM: out[N,C] = A[N,128] * W[128,C] ----------------
// One block = 16 rows; 8 waves; wave w computes column tile [w*16, w*16+16).
template<int C>
__global__ __launch_bounds__(256) void k_gemm_wmma(const float* __restrict__ A,
                                                   const float* __restrict__ W,
                                                   float* __restrict__ out, int N) {
    __shared__ float As[16][33];      // stride 33: distinct banks for lanes 0..15
    __shared__ float Bs[32][C];

    const int tid  = threadIdx.x;
    const int lane = tid & 31;
    const int wv   = tid >> 5;
    const int half = lane >> 4;       // 0: lanes 0-15, 1: lanes 16-31
    const int r    = lane & 15;
    const int m0   = blockIdx.x * 16;
    const int n0   = wv * 16;

    v8f acc = {};

    for (int k0 = 0; k0 < 128; k0 += 32) {
        // stage A tile 16x32 (512 elems, 2 per thread, coalesced float2 reads)
        {
            int idx = tid * 2;
            int row = idx >> 5;
            int col = idx & 31;
            float v0 = 0.0f, v1 = 0.0f;
            int gr = m0 + row;
            if (gr < N) {
                const float* p = A + (size_t)gr * 128 + k0 + col;
                v0 = p[0];
                v1 = p[1];
            }
            As[row][col]     = v0;
            As[row][col + 1] = v1;
        }
        // stage B tile 32xC (coalesced)
        for (int idx = tid; idx < 32 * C; idx += 256) {
            int row = idx / C, col = idx % C;
            Bs[row][col] = W[(size_t)(k0 + row) * C + col];
        }
        __syncthreads();

        if (n0 < C) {
#pragma unroll
            for (int kk = 0; kk < 32; kk += 4) {
                v2f a, b;
                // A 16x4: lanes 0-15 -> K=kk,kk+1 ; lanes 16-31 -> K=kk+2,kk+3
                a.x = As[r][kk + 2 * half];
                a.y = As[r][kk + 2 * half + 1];
                // B 4x16: VGPR0 = row K (lower lanes) / K+2 (upper), VGPR1 = K+1 / K+3
                b.x = Bs[kk + 2 * half][n0 + r];
                b.y = Bs[kk + 2 * half + 1][n0 + r];
                acc = __builtin_amdgcn_wmma_f32_16x16x4_f32(
                    /*neg_a=*/false, a, /*neg_b=*/false, b,
                    /*c_mod=*/(short)0, acc, /*reuse_a=*/false, /*reuse_b=*/false);
            }
        }
        __syncthreads();
    }

    if (n0 < C) {
#pragma unroll
        for (int r2 = 0; r2 < 8; ++r2) {
            int row = m0 + r2 + 8 * half;     // D: VGPR r2 -> M=r2 (lo lanes) / r2+8 (hi)
            if (row < N) out[(size_t)row * C + n0 + r] = acc[r2];
        }
    }
}

// ---------------- edge aggregation: one wave per message ----------------
// messages 0..E-1 are edges, E..E+N-1 are self loops.
template<int C>
__global__ void k_aggregate(const float* __restrict__ feat,
                            const int* __restrict__ src,
                            const int* __restrict__ dst,
                            const float* __restrict__ dinv,
                            float* out, int E, int N) {
    int gw   = (blockIdx.x * blockDim.x + threadIdx.x) >> 5;
    int lane = threadIdx.x & 31;
    int total = E + N;
    if (gw >= total) return;

    int s, d;
    if (gw < E) { s = src[gw]; d = dst[gw]; }
    else        { s = d = gw - E; }

    float w = dinv[s] * dinv[d];

    if (C == 128) {
        const float4* fp = (const float4*)(feat + (size_t)s * C);
        float4 v = fp[lane];
        float* op = out + (size_t)d * C + lane * 4;
        atomicAdd(op + 0, v.x * w);
        atomicAdd(op + 1, v.y * w);
        atomicAdd(op + 2, v.z * w);
        atomicAdd(op + 3, v.w * w);
    } else {
        const float2* fp = (const float2*)(feat + (size_t)s * C);
        float2 v = fp[lane];
        float* op = out + (size_t)d * C + lane * 2;
        atomicAdd(op + 0, v.x * w);
        atomicAdd(op + 1, v.y * w);
    }
}

// ---------------- elementwise ----------------
__global__ void k_bias_relu(float* h, const float* __restrict__ b, int total, int C) {
    int i = blockIdx.x * blockDim.x + threadIdx.x;
    if (i < total) {
        float v = h[i] + b[i % C];
        h[i] = v > 0.0f ? v : 0.0f;
    }
}

__global__ void k_bias_set(float* o, const float* __restrict__ b, int total, int C) {
    int i = blockIdx.x * blockDim.x + threadIdx.x;
    if (i < total) o[i] = b[i % C];
}

// ---------------- launch ----------------
extern "C" void kernel_launch(void* const* d_in, const int* in_sizes, int n_in,
                              void* d_out, int out_size, void* d_ws, size_t ws_size,
                              hipStream_t stream) {
    const float* x  = (const float*)d_in[0];
    const int*   ei = (const int*)  d_in[1];
    const float* W1 = (const float*)d_in[2];
    const float* b1 = (const float*)d_in[3];
    const float* W2 = (const float*)d_in[4];
    const float* b2 = (const float*)d_in[5];

    const int N = in_sizes[0] / 128;
    const int E = in_sizes[1] / 2;
    const int* src = ei;
    const int* dst = ei + E;

    char* ws = (char*)d_ws;
    float* dinv = (float*)ws;                                   // N floats
    size_t off  = (((size_t)N * 4) + 255) & ~(size_t)255;
    float* bufA = (float*)(ws + off);                            // N*128 floats
    size_t szA  = (size_t)N * 128 * sizeof(float);
    float* bufB = (float*)(ws + off + szA);                      // N*128 floats

    const int msgs = E + N;

    // normalization coefficients
    k_init_deg<<<(N + 255) / 256, 256, 0, stream>>>(dinv, N);
    k_count_deg<<<(E + 255) / 256, 256, 0, stream>>>(dst, dinv, E);
    k_rsqrt_inplace<<<(N + 255) / 256, 256, 0, stream>>>(dinv, N);

    // layer 1: xW1
    k_gemm_wmma<128><<<(N + 15) / 16, 256, 0, stream>>>(x, W1, bufA, N);

    // layer 1: normalized scatter-add (agg buffer zeroed every call)
    hipMemsetAsync(bufB, 0, szA, stream);
    {
        long long th = (long long)msgs * 32;
        k_aggregate<128><<<(int)((th + 255) / 256), 256, 0, stream>>>(bufA, src, dst, dinv, bufB, E, N);
    }
    k_bias_relu<<<(N * 128 + 255) / 256, 256, 0, stream>>>(bufB, b1, N * 128, 128);

    // layer 2: hW2 (reuse bufA, needs only N*64 floats)
    k_gemm_wmma<64><<<(N + 15) / 16, 256, 0, stream>>>(bufB, W2, bufA, N);

    // layer 2: out = b2 + scatter-add
    k_bias_set<<<(N * 64 + 255) / 256, 256, 0, stream>>>((float*)d_out, b2, N * 64, 64);
    {
        long long th = (long long)msgs * 32;
        k_aggregate<64><<<(int)((th + 255) / 256), 256, 0, stream>>>(bufA, src, dst, dinv, (float*)d_out, E, N);
    }
}